// OrthogonalTransform_21165598835192
// MI455X (gfx1250) — compile-verified
//
#include <hip/hip_runtime.h>

typedef __attribute__((ext_vector_type(2))) float v2f;
typedef __attribute__((ext_vector_type(4))) float v4f;
typedef __attribute__((ext_vector_type(8))) float v8f;

namespace {
constexpr int kN     = 2048;            // transform dim (== M, reflector count)
constexpr int kB     = 4096;            // batch
constexpr int kCols  = 16;              // batch columns per workgroup
constexpr int kBlk   = 16;              // reflectors per compact-WY block
constexpr int kWaves = 8;
constexpr int kThreads = kWaves * 32;
constexpr int kHStr  = kN + 4;          // LDS row stride (pad -> 4-bank skew)
constexpr int kSmemFloats = kCols * kHStr   // h slab (transposed)
                          + 2 * kWaves * 256 // W / G partials per wave
                          + 4 * 256;         // Wbuf, Gbuf, Tbuf, Ybuf
}

__device__ __forceinline__ v8f wmma_f32_16x16x4(v2f a, v2f b, v8f c) {
  // D = A(16x4, f32) * B(4x16, f32) + C(16x16, f32)
  return __builtin_amdgcn_wmma_f32_16x16x4_f32(
      /*neg_a=*/false, a, /*neg_b=*/false, b,
      /*c_mod=*/(short)0, c, /*reuse_a=*/false, /*reuse_b=*/false);
}

// Element (row, col) of U = triu(weight) with U[N-1][N-1] = 1.0 (H1).
// Only used on peeled boundary tiles; hot loops use raw b64 loads.
__device__ __forceinline__ float load_u(const float* __restrict__ w, int row, int col) {
  float v = 0.0f;
  if (col >= row) v = w[(size_t)row * kN + col];
  if (row == kN - 1 && col == kN - 1) v = 1.0f;
  return v;
}

__global__ __launch_bounds__(kThreads)
void ortho_householder_kernel(const float* __restrict__ x,
                              const float* __restrict__ w,
                              float* __restrict__ out) {
  extern __shared__ float smem[];
  float* hbuf  = smem;                      // [kCols][kHStr], hbuf[c][k] = h[k][c]
  float* Wpart = hbuf + kCols * kHStr;      // [kWaves][16*16] partial V^T h
  float* Gpart = Wpart + kWaves * 256;      // [kWaves][16*16] partial V^T V
  float* Wbuf  = Gpart + kWaves * 256;      // [16][16]
  float* Gbuf  = Wbuf + 256;                // [16][16]
  float* Tbuf  = Gbuf + 256;                // [16][16] compact-WY T
  float* Ybuf  = Tbuf + 256;                // [16][16] = -T * W

  const int tid  = threadIdx.x;
  // Provably wave-uniform -> scalar loop counters, no EXEC-masked loops.
  const int wave = __builtin_amdgcn_readfirstlane(tid >> 5);
  const int lane = tid & 31;
  const int lm   = lane & 15;
  const int lh   = lane >> 4;               // half-wave: selects K/M sub-range
  const int c0   = blockIdx.x * kCols;
  float* const hrow = hbuf + lm * kHStr;    // this lane's batch-column row

  // ---- Load h slab transposed into LDS: hbuf[c][k] = x[(c0+c)*N + k]
  for (int i = tid; i < kCols * (kN / 4); i += kThreads) {
    const int c  = i / (kN / 4);
    const int k4 = (i - c * (kN / 4)) * 4;
    v4f v = *(const v4f*)(x + (size_t)(c0 + c) * kN + k4);
    *(v4f*)(hbuf + c * kHStr + k4) = v;
  }
  __syncthreads();

  // Reflectors applied i = N-1 .. 0  =>  blocks base = N-16 down to 0.
  // Within a block, Q = H_base ... H_{base+15} = I - V T V^T.
  for (int base = kN - kBlk; base >= 0; base -= kBlk) {
    // ---- Phase 1: W = V^T h (16x16) and G = V^T V (16x16), K split over waves.
    // Triangular mask (col >= row) and the H1 substitute only affect
    // k in [base, base+16) == the first k-tile of waves 0..3: peel it.
    v8f accW = {};
    v8f accG = {};
    int k = base + wave * 4;
    if (wave < 4) {                    // wave-uniform; EXEC full at the WMMAs
      const int col = k + 2 * lh;
      const int row = base + lm;
      v2f a;
      a.x = load_u(w, row, col);
      a.y = load_u(w, row, col + 1);
      v2f b = *(const v2f*)(hrow + col);
      accW = wmma_f32_16x16x4(a, b, accW);
      accG = wmma_f32_16x16x4(a, a, accG);
      k += kWaves * 4;
    }
    // Unmasked main reduction, software-pipelined: issue iteration i+1's
    // loads before consuming iteration i's registers in the WMMAs so the
    // waits become counted (overlap one full load latency with the MMAs).
    if (k < kN) {                      // scalar-provable guard (k wave-uniform)
      const float* wrow = w + (size_t)(base + lm) * kN;  // A row for this lane
      v2f a = *(const v2f*)(wrow + k + 2 * lh);
      v2f b = *(const v2f*)(hrow + k + 2 * lh);
#pragma unroll 2
      for (k += kWaves * 4; k < kN; k += kWaves * 4) {
        const int col = k + 2 * lh;
        v2f an = *(const v2f*)(wrow + col);   // unmasked: col >= base+16 > row
        v2f bn = *(const v2f*)(hrow + col);
        accW = wmma_f32_16x16x4(a, b, accW);
        accG = wmma_f32_16x16x4(a, a, accG);  // Gram: same regs = B-layout A^T
        a = an;
        b = bn;
      }
      accW = wmma_f32_16x16x4(a, b, accW);    // drain pipeline
      accG = wmma_f32_16x16x4(a, a, accG);
    }
#pragma unroll
    for (int r = 0; r < 8; ++r) {   // D layout: VGPR r, halves -> M = r + 8*lh
      const int m = r + 8 * lh;
      Wpart[wave * 256 + m * 16 + lm] = accW[r];
      Gpart[wave * 256 + m * 16 + lm] = accG[r];
    }
    __syncthreads();

    // ---- Reduce wave partials (256 threads, one element each)
    {
      float sw = 0.f, sg = 0.f;
#pragma unroll
      for (int p = 0; p < kWaves; ++p) {
        sw += Wpart[p * 256 + tid];
        sg += Gpart[p * 256 + tid];
      }
      Wbuf[tid] = sw;
      Gbuf[tid] = sg;
    }
    __syncthreads();

    // ---- T recurrence: lane i owns row i, fully unrolled (registers only).
    // c_j = 2/||u_j||^2 (0 if u_j == 0 -> reflector skipped), recomputed
    // per lane from Gbuf's diagonal to avoid an extra barrier.
    if (tid < 16) {
      const int i = tid;
      float Trow[16];
#pragma unroll
      for (int j = 0; j < 16; ++j) {
        const float gd = Gbuf[j * 16 + j];
        const float cj = (gd > 0.f) ? (2.0f / gd) : 0.0f;
        float s = 0.f;
#pragma unroll
        for (int l = 0; l < 16; ++l) {
          if (l < j) s += (l >= i) ? Trow[l] * Gbuf[l * 16 + j] : 0.f;
        }
        float tv = 0.f;
        if (j == i) tv = cj;
        else if (j > i) tv = -cj * s;
        Trow[j] = tv;
        Tbuf[i * 16 + j] = tv;
      }
    }
    __syncthreads();

    // ---- Y = -(T * W): tiny 16x16x16, one element per thread
    {
      const int m = tid >> 4;
      const int c = tid & 15;
      float s = 0.f;
#pragma unroll
      for (int l = 0; l < 16; ++l) s += Tbuf[m * 16 + l] * Wbuf[l * 16 + c];
      Ybuf[m * 16 + c] = -s;
    }
    __syncthreads();

    // ---- Phase 3: h += V * Y  (i.e. h -= V T V^T h), row-tiles over waves.
    // Only tile t == 0 (k0 == base) touches the triangular/H1 region.
    const int ntiles = (kN - base) >> 4;
    v2f by[4];
#pragma unroll
    for (int m0 = 0; m0 < 16; m0 += 4) {      // B operand is tile-invariant
      by[m0 >> 2].x = Ybuf[(m0 + 2 * lh) * 16 + lm];
      by[m0 >> 2].y = Ybuf[(m0 + 2 * lh + 1) * 16 + lm];
    }
    for (int t = wave; t < ntiles; t += kWaves) {
      const int k0 = base + t * 16;
      float* hp = hrow + k0 + 8 * lh;         // C/D tile, rows along k
      v4f clo = *(const v4f*)(hp);
      v4f chi = *(const v4f*)(hp + 4);
      v8f acc = {clo.x, clo.y, clo.z, clo.w, chi.x, chi.y, chi.z, chi.w};
      if (t == 0) {                            // wave-uniform masked tile
#pragma unroll
        for (int m0 = 0; m0 < 16; m0 += 4) {
          v2f a;  // A[r][mm] = V[k0+r][m0+mm] = U[base+m0+mm][k0+r]
          a.x = load_u(w, base + m0 + 2 * lh,     k0 + lm);
          a.y = load_u(w, base + m0 + 2 * lh + 1, k0 + lm);
          acc = wmma_f32_16x16x4(a, by[m0 >> 2], acc);
        }
      } else {                                 // k0 >= base+16: unmasked
        const float* wp = w + (size_t)(base + 2 * lh) * kN + k0 + lm;
#pragma unroll
        for (int m0 = 0; m0 < 16; m0 += 4) {
          v2f a;
          a.x = wp[(size_t)m0 * kN];
          a.y = wp[(size_t)(m0 + 1) * kN];
          acc = wmma_f32_16x16x4(a, by[m0 >> 2], acc);
        }
      }
      v4f olo = {acc[0], acc[1], acc[2], acc[3]};
      v4f ohi = {acc[4], acc[5], acc[6], acc[7]};
      *(v4f*)(hp) = olo;
      *(v4f*)(hp + 4) = ohi;
    }
    __syncthreads();
  }

  // ---- Write result: out[(c0+c)*N + k] = hbuf[c][k]
  for (int i = tid; i < kCols * (kN / 4); i += kThreads) {
    const int c  = i / (kN / 4);
    const int k4 = (i - c * (kN / 4)) * 4;
    v4f v = *(const v4f*)(hbuf + c * kHStr + k4);
    *(v4f*)(out + (size_t)(c0 + c) * kN + k4) = v;
  }
}

extern "C" void kernel_launch(void* const* d_in, const int* in_sizes, int n_in,
                              void* d_out, int out_size, void* d_ws, size_t ws_size,
                              hipStream_t stream) {
  const float* x = (const float*)d_in[0];   // (B, N) fp32
  const float* w = (const float*)d_in[1];   // (N, M) fp32
  float* out = (float*)d_out;               // (B, N) fp32

  dim3 grid(kB / kCols);                    // 256 workgroups
  dim3 block(kThreads);                     // 256 threads = 8 wave32
  const size_t smem = (size_t)kSmemFloats * sizeof(float);  // ~151 KB / WG
  hipLaunchKernelGGL(ortho_householder_kernel, grid, block, smem, stream,
                     x, w, out);
}